// AttentionFusion_8177617731865
// MI455X (gfx1250) — compile-verified
//
#include <hip/hip_runtime.h>
#include <stdint.h>

// Problem constants (B,N,C,H,W) = (8,2,512,64,64)
#define NBATCH   8
#define NCH      512                  // C
#define HW       4096                 // H*W
#define MELEMS   (NCH * HW)           // 2,097,152 per (b,n) row
#define KDIM     (2 * NCH)            // 1024 (GEMM K)
#define ODIM     NCH                  // 512  (GEMM M / out channels)
#define PDIM     HW                   // 4096 (GEMM N / pixels)
#define EPSV     1e-5f

#define RED_BLOCKS 512
#define RED_THREADS 256

typedef __attribute__((ext_vector_type(2))) float v2f;
typedef __attribute__((ext_vector_type(8))) float v8f;

// ---------------------------------------------------------------------------
// CDNA5 async copy: global -> LDS, 16 bytes per lane, tracked by ASYNCcnt.
// ---------------------------------------------------------------------------
__device__ __forceinline__ void async_copy_b128(uint32_t lds_off, const void* gaddr)
{
    asm volatile("global_load_async_to_lds_b128 %0, %1, off"
                 :: "v"(lds_off), "v"(gaddr)
                 : "memory");
}

__device__ __forceinline__ void wait_asynccnt0()
{
    asm volatile("s_wait_asynccnt 0x0" ::: "memory");
}

// ---------------------------------------------------------------------------
// Kernel 1: per-batch streaming reduction of 5 quantities:
//   S0 = sum x0, S1 = sum x1, S00 = sum x0*x0, S01 = sum x0*x1, S11 = sum x1*x1
// Fixed-order partials (no float atomics) -> bitwise deterministic.
// ---------------------------------------------------------------------------
__global__ __launch_bounds__(RED_THREADS)
void af_reduce5_kernel(const float* __restrict__ x, float* __restrict__ partial)
{
    const int b = blockIdx.y;
    const float4* __restrict__ x0 =
        (const float4*)(x + (size_t)b * 2 * MELEMS);
    const float4* __restrict__ x1 =
        (const float4*)(x + (size_t)b * 2 * MELEMS + MELEMS);

    const int tid = threadIdx.x;
    const int gid = blockIdx.x * RED_THREADS + tid;
    const int stride = gridDim.x * RED_THREADS;

    float s0 = 0.f, s1 = 0.f, s00 = 0.f, s01 = 0.f, s11 = 0.f;
    for (int i = gid; i < MELEMS / 4; i += stride) {
        float4 a = x0[i];
        float4 c = x1[i];
        s0  += a.x + a.y + a.z + a.w;
        s1  += c.x + c.y + c.z + c.w;
        s00 += a.x * a.x + a.y * a.y + a.z * a.z + a.w * a.w;
        s01 += a.x * c.x + a.y * c.y + a.z * c.z + a.w * c.w;
        s11 += c.x * c.x + c.y * c.y + c.z * c.z + c.w * c.w;
    }

    __shared__ float sm[RED_THREADS];
    float vals[5] = { s0, s1, s00, s01, s11 };
#pragma unroll
    for (int q = 0; q < 5; ++q) {
        sm[tid] = vals[q];
        __syncthreads();
        for (int off = RED_THREADS / 2; off > 0; off >>= 1) {
            if (tid < off) sm[tid] += sm[tid + off];
            __syncthreads();
        }
        if (tid == 0)
            partial[((size_t)b * RED_BLOCKS + blockIdx.x) * 5 + q] = sm[0];
        __syncthreads();
    }
}

// ---------------------------------------------------------------------------
// Kernel 2: one block per batch. Final reduce + cov -> RMS-norm -> tanh,
// attp[n][k] = att[n][k] + (n==k)  (residual folded in).
// ---------------------------------------------------------------------------
__global__ __launch_bounds__(256)
void af_attention_kernel(const float* __restrict__ partial,
                         const float* __restrict__ alpha,
                         const float* __restrict__ gamma,
                         const float* __restrict__ beta,
                         float* __restrict__ attp)
{
    const int b = blockIdx.x;
    const int tid = threadIdx.x;
    __shared__ float sm[256];
    __shared__ float res[5];

#pragma unroll
    for (int q = 0; q < 5; ++q) {
        float v = partial[((size_t)b * RED_BLOCKS + tid) * 5 + q]
                + partial[((size_t)b * RED_BLOCKS + tid + 256) * 5 + q];
        sm[tid] = v;
        __syncthreads();
        for (int off = 128; off > 0; off >>= 1) {
            if (tid < off) sm[tid] += sm[tid + off];
            __syncthreads();
        }
        if (tid == 0) res[q] = sm[0];
        __syncthreads();
    }

    if (tid == 0) {
        const float Mf = (float)MELEMS;
        const float sc = alpha[0] / (Mf - 1.0f);
        float S0 = res[0], S1 = res[1];
        float c00 = (res[2] - S0 * S0 / Mf) * sc;
        float c01 = (res[3] - S0 * S1 / Mf) * sc;
        float c11 = (res[4] - S1 * S1 / Mf) * sc;
        float ms = (c00 * c00 + 2.0f * c01 * c01 + c11 * c11) * 0.25f;
        float inv = 1.0f / sqrtf(ms + EPSV);
        float g = gamma[0], bt = beta[0];
        float a00 = tanhf(g * (c00 * inv) + bt);
        float a01 = tanhf(g * (c01 * inv) + bt);
        float a11 = tanhf(g * (c11 * inv) + bt);
        attp[b * 4 + 0] = a00 + 1.0f; // n=0,k=0
        attp[b * 4 + 1] = a01;        // n=0,k=1
        attp[b * 4 + 2] = a01;        // n=1,k=0
        attp[b * 4 + 3] = a11 + 1.0f; // n=1,k=1
    }
}

// ---------------------------------------------------------------------------
// Kernel 3: fold attention+residual into the conv weight:
//   Weff[b][o][k*C + c] = W[o][c] * attp[b][0][k] + W[o][C + c] * attp[b][1][k]
// ---------------------------------------------------------------------------
__global__ __launch_bounds__(256)
void af_weff_kernel(const float* __restrict__ W,
                    const float* __restrict__ attp,
                    float* __restrict__ weff)
{
    const int idx = blockIdx.x * 256 + threadIdx.x;  // [0, 8*512*1024)
    const int b   = idx >> 19;
    const int rem = idx & ((1 << 19) - 1);
    const int o   = rem >> 10;
    const int kc  = rem & 1023;
    const int k   = kc >> 9;
    const int c   = kc & 511;
    const float a0 = attp[b * 4 + k];
    const float a1 = attp[b * 4 + 2 + k];
    weff[idx] = W[o * KDIM + c] * a0 + W[o * KDIM + NCH + c] * a1;
}

// ---------------------------------------------------------------------------
// Kernel 4: per-batch fp32 WMMA GEMM: out[b] = Weff[b](512x1024) @ x[b](1024x4096)
// Block = 256 threads (8 wave32, 2x4 grid), 64x256 block tile, K-chunk = 16.
// Wave tile 32x64: 2 M-frags x 4 N-frags = 8 f32 accumulators.
// Double-buffered LDS: B tile prefetched via global_load_async_to_lds_b128
// (ASYNCcnt waited only AFTER the chunk's 32 WMMAs); A tile stored
// pair-interleaved (v2f) so each A fragment is one aligned ds_load_b64.
// ---------------------------------------------------------------------------
__global__ __launch_bounds__(256)
void af_gemm_wmma_kernel(const float* __restrict__ x,
                         const float* __restrict__ weff,
                         float* __restrict__ out)
{
    const int b  = blockIdx.z;
    const int p0 = blockIdx.x * 256;
    const int o0 = blockIdx.y * 64;

    const float* __restrict__ A = weff + (size_t)b * ODIM * KDIM;  // [512][1024]
    const float* __restrict__ B = x    + (size_t)b * KDIM * PDIM;  // [1024][4096]
    float* __restrict__ C       = out  + (size_t)b * ODIM * PDIM;  // [512][4096]

    // As2[buf][kp][row] = { A[2kp][row], A[2kp+1][row] }  (K-pair interleaved)
    __shared__ v2f   As2[2][8][64];     // 2 x 4 KB
    __shared__ float Bs [2][16][256];   // 2 x 16 KB

    const int tid  = threadIdx.x;
    const int lane = tid & 31;
    const int wave = tid >> 5;
    const int wm   = wave >> 2;      // 0..1  (32-row slice)
    const int wn   = wave & 3;       // 0..3  (64-col slice)
    const int l    = lane & 15;
    const int hi   = (lane >> 4) & 1;

    v8f acc[2][4];
#pragma unroll
    for (int mf = 0; mf < 2; ++mf)
#pragma unroll
        for (int nf = 0; nf < 4; ++nf)
            acc[mf][nf] = v8f{};

    // --- A staging map: one float4 per thread -> two v2f LDS stores ---
    const int arow = tid & 63;            // 0..63 (distinct banks per wave)
    const int akp  = (tid >> 6) * 2;      // K-pair index: 0,2,4,6
    const float* agptr = A + (size_t)arow * KDIM + akp * 2;

    // --- B staging map: four async b128 per thread, per buffer ---
    uint32_t blds[2][4];
    const float* bgp[4];
#pragma unroll
    for (int t = 0; t < 4; ++t) {
        const int slot = tid + t * 256;   // 0..1023 chunk slots of 16B
        const int row  = slot >> 6;       // 0..15
        const int cf   = (slot & 63) * 4; // 0..252
        blds[0][t] = (uint32_t)(uintptr_t)&Bs[0][row][cf];
        blds[1][t] = (uint32_t)(uintptr_t)&Bs[1][row][cf];
        bgp[t]     = B + (size_t)row * PDIM + p0 + cf;
    }

    // ---- prologue: stage chunk 0 into buffer 0 ----
    {
        float4 av = *(const float4*)agptr;
        agptr += 16;
        As2[0][akp + 0][arow] = v2f{av.x, av.y};
        As2[0][akp + 1][arow] = v2f{av.z, av.w};
#pragma unroll
        for (int t = 0; t < 4; ++t) {
            async_copy_b128(blds[0][t], bgp[t]);
            bgp[t] += 16 * PDIM;
        }
        wait_asynccnt0();
        __syncthreads();
    }

    for (int kc = 0; kc < KDIM; kc += 16) {
        const int  cur  = (kc >> 4) & 1;
        const int  nxt  = cur ^ 1;
        const bool more = (kc + 16) < KDIM;

        // prefetch next chunk: A -> regs (consumed after compute),
        // B -> LDS via async DMA (waited after compute)
        float4 av;
        if (more) {
            av = *(const float4*)agptr;
            agptr += 16;
#pragma unroll
            for (int t = 0; t < 4; ++t) {
                async_copy_b128(blds[nxt][t], bgp[t]);
                bgp[t] += 16 * PDIM;
            }
        }

        // ---- compute on current buffer ----
#pragma unroll
        for (int ks = 0; ks < 16; ks += 4) {
            const int kp = ks >> 1;               // pair index 0,2,4,6
            v2f af[2];
#pragma unroll
            for (int mf = 0; mf < 2; ++mf) {
                const int r = wm * 32 + mf * 16 + l;
                af[mf] = As2[cur][kp + hi][r];    // one ds_load_b64
            }
#pragma unroll
            for (int nf = 0; nf < 4; ++nf) {
                const int col = wn * 64 + nf * 16 + l;
                v2f bf;
                bf.x = Bs[cur][ks + (hi ? 2 : 0)][col];
                bf.y = Bs[cur][ks + (hi ? 3 : 1)][col];
#pragma unroll
                for (int mf = 0; mf < 2; ++mf) {
                    acc[mf][nf] = __builtin_amdgcn_wmma_f32_16x16x4_f32(
                        false, af[mf], false, bf,
                        (short)0, acc[mf][nf], false, false);
                }
            }
        }

        // finish staging next buffer, then block-wide handoff
        if (more) {
            As2[nxt][akp + 0][arow] = v2f{av.x, av.y};
            As2[nxt][akp + 1][arow] = v2f{av.z, av.w};
            wait_asynccnt0();
        }
        __syncthreads();
    }

    // D layout: VGPR v holds row v (lanes 0-15) and row v+8 (lanes 16-31)
#pragma unroll
    for (int mf = 0; mf < 2; ++mf) {
#pragma unroll
        for (int nf = 0; nf < 4; ++nf) {
            const int col = p0 + wn * 64 + nf * 16 + l;
#pragma unroll
            for (int v = 0; v < 8; ++v) {
                const int row = o0 + wm * 32 + mf * 16 + v + (hi ? 8 : 0);
                C[(size_t)row * PDIM + col] = acc[mf][nf][v];
            }
        }
    }
}

// ---------------------------------------------------------------------------
extern "C" void kernel_launch(void* const* d_in, const int* in_sizes, int n_in,
                              void* d_out, int out_size, void* d_ws, size_t ws_size,
                              hipStream_t stream)
{
    const float* x      = (const float*)d_in[0];
    const float* alpha  = (const float*)d_in[1];
    const float* gamma  = (const float*)d_in[2];
    const float* beta   = (const float*)d_in[3];
    const float* conv_w = (const float*)d_in[4];
    float* out = (float*)d_out;

    float* ws      = (float*)d_ws;
    float* partial = ws;                                  // 8*512*5 = 20480 floats
    float* attp    = ws + NBATCH * RED_BLOCKS * 5;        // 32 floats
    float* weff    = ws + NBATCH * RED_BLOCKS * 5 + 32;   // 8*512*1024 floats, 16B-aligned

    af_reduce5_kernel<<<dim3(RED_BLOCKS, NBATCH), RED_THREADS, 0, stream>>>(x, partial);
    af_attention_kernel<<<NBATCH, 256, 0, stream>>>(partial, alpha, gamma, beta, attp);
    af_weff_kernel<<<(NBATCH * ODIM * KDIM) / 256, 256, 0, stream>>>(conv_w, attp, weff);
    af_gemm_wmma_kernel<<<dim3(PDIM / 256, ODIM / 64, NBATCH), 256, 0, stream>>>(x, weff, out);
}